// RulesLoss_46961172414632
// MI455X (gfx1250) — compile-verified
//
#include <hip/hip_runtime.h>

typedef __attribute__((ext_vector_type(2))) float v2f;
typedef __attribute__((ext_vector_type(8))) float v8f;

#define T_DIM 8192
#define B_DIM 1024
#define EPT   8                         // elements per thread (row-aligned chunk)
#define BLOCK 256
#define NWAVE (BLOCK / 32)
#define NBLK  4096                      // (B*T)/EPT/BLOCK
#define RED_BLOCK 256
#define RED_PER   (NBLK / RED_BLOCK)    // 16

// ---- table lookups collapsed to 64-bit immediates ----
// mask1 keys: 27,43 | 92,94,100,122    mask2 keys: 27,28,29,43,44 | 121
#define M1_LO ((1ULL << 27) | (1ULL << 43))
#define M1_HI ((1ULL << (92-64)) | (1ULL << (94-64)) | (1ULL << (100-64)) | (1ULL << (122-64)))
#define M2_LO ((1ULL << 27) | (1ULL << 28) | (1ULL << 29) | (1ULL << 43) | (1ULL << 44))
#define M2_HI (1ULL << (121-64))

__device__ __forceinline__ float gap_core(float dp, float dpn, int tok, bool has_next) {
    unsigned long long w1 = (tok < 64) ? M1_LO : M1_HI;
    unsigned long long w2 = (tok < 64) ? M2_LO : M2_HI;
    bool in1 = (w1 >> (tok & 63)) & 1ULL;
    bool in2 = (w2 >> (tok & 63)) & 1ULL;
    // expected only consumed when in1: {27,43}->5, {92,94,100}->2, {122}->3
    float expected = (tok == 122) ? 3.0f : ((tok < 64) ? 5.0f : 2.0f);
    float gap1 = (in1 && dp > expected) ? (dp - expected) : 0.0f;
    float gap2 = (in2 && has_next && (3.0f * dp > dpn)) ? (dp - dpn * (1.0f / 3.0f)) : 0.0f;
    return gap1 + gap2;
}

// Wave-wide sum via V_WMMA_F32_16X16X4_F32:
// A vgpr0 = acc (A[m][0]=lane m, A[m][2]=lane m+16), A vgpr1 = 0, B = ones
// => D[m][n] = acc[m] + acc[m+16]; per-lane sum of C rows + shfl_xor(16) = wave total.
__device__ __forceinline__ float wave_reduce_wmma(float acc) {
    v2f a; a[0] = acc;  a[1] = 0.0f;
    v2f b; b[0] = 1.0f; b[1] = 1.0f;
    v8f c = {};
    c = __builtin_amdgcn_wmma_f32_16x16x4_f32(false, a, false, b, (short)0, c, false, false);
    float s = c[0] + c[1] + c[2] + c[3] + c[4] + c[5] + c[6] + c[7];
    s += __shfl_xor(s, 16, 32);
    return s;  // full wave total in every lane
}

__global__ void __launch_bounds__(BLOCK)
rules_loss_kernel(const float* __restrict__ dur,
                  const int*   __restrict__ tok2,   // int64 tokens viewed as int pairs
                  float*       __restrict__ partial) {
    const int tid = blockIdx.x * BLOCK + threadIdx.x;
    const long long base = (long long)tid * EPT;
    const int t = (int)(base & (T_DIM - 1));        // position within row

    // dur_pred: 2x B128 loads
    const float4* dp4p = (const float4*)(dur + base);
    float4 d0 = dp4p[0];
    float4 d1 = dp4p[1];
    float dp[10];
    dp[0] = d0.x; dp[1] = d0.y; dp[2] = d0.z; dp[3] = d0.w;
    dp[4] = d1.x; dp[5] = d1.y; dp[6] = d1.z; dp[7] = d1.w;

    // tokens: 4x B128 loads, keep low dword of each int64
    int tk[9];
    const int4* tkp = (const int4*)(tok2 + 2 * base);
#pragma unroll
    for (int q = 0; q < 4; ++q) {
        int4 v = tkp[q];
        tk[2 * q]     = v.x;
        tk[2 * q + 1] = v.z;
    }

    float acc = 0.0f;
    if (t != T_DIM - EPT) {
        // Fast path (1023/1024 threads): all neighbors exist, no guards anywhere.
        dp[8] = dur[base + 8];
        dp[9] = dur[base + 9];
        tk[8] = tok2[2 * (base + 8)];
        float g[9];
#pragma unroll
        for (int j = 0; j < 9; ++j)
            g[j] = gap_core(dp[j], dp[j + 1], tk[j], true);
#pragma unroll
        for (int j = 0; j < 8; ++j) {
            float dr  = dp[j] - g[j] + g[j + 1];
            float dlt = __logf(dp[j] + 1.0f) - __logf(dr + 1.0f);
            acc += dlt * dlt;
        }
    } else {
        // Row-end chunk: guarded version.
        dp[8] = 0.0f; dp[9] = 0.0f; tk[8] = 0;
        float g[9];
#pragma unroll
        for (int j = 0; j < 9; ++j)
            g[j] = gap_core(dp[j], dp[j + 1], tk[j], (t + j) < (T_DIM - 1));
#pragma unroll
        for (int j = 0; j < 8; ++j) {
            float gnext = ((t + j) < (T_DIM - 1)) ? g[j + 1] : 0.0f;
            float dr  = dp[j] - g[j] + gnext;
            float dlt = __logf(dp[j] + 1.0f) - __logf(dr + 1.0f);
            acc += dlt * dlt;
        }
    }

    // Paths reconverge here: EXEC all-1s for the WMMA reduction.
    float s = wave_reduce_wmma(acc);

    __shared__ float wsum[NWAVE];
    if ((threadIdx.x & 31) == 0) wsum[threadIdx.x >> 5] = s;
    __syncthreads();
    if (threadIdx.x == 0) {
        float tot = 0.0f;
#pragma unroll
        for (int w = 0; w < NWAVE; ++w) tot += wsum[w];
        partial[blockIdx.x] = tot;      // no atomics anywhere
    }
}

__global__ void __launch_bounds__(RED_BLOCK)
final_reduce_kernel(const float* __restrict__ partial, float* __restrict__ out) {
    float acc = 0.0f;
#pragma unroll
    for (int i = 0; i < RED_PER; ++i)
        acc += partial[threadIdx.x + RED_BLOCK * i];   // coalesced, strided

    float s = wave_reduce_wmma(acc);

    __shared__ float wsum[RED_BLOCK / 32];
    if ((threadIdx.x & 31) == 0) wsum[threadIdx.x >> 5] = s;
    __syncthreads();
    if (threadIdx.x == 0) {
        float tot = 0.0f;
#pragma unroll
        for (int w = 0; w < RED_BLOCK / 32; ++w) tot += wsum[w];
        out[0] = tot * (0.6f / ((float)B_DIM * (float)T_DIM));  // SCALE / N
    }
}

extern "C" void kernel_launch(void* const* d_in, const int* in_sizes, int n_in,
                              void* d_out, int out_size, void* d_ws, size_t ws_size,
                              hipStream_t stream) {
    const float* dur  = (const float*)d_in[0];
    const int*   tok2 = (const int*)d_in[1];   // int64 tokens read as int pairs
    float* out      = (float*)d_out;
    float* partials = (float*)d_ws;            // NBLK floats = 16 KB scratch

    rules_loss_kernel<<<NBLK, BLOCK, 0, stream>>>(dur, tok2, partials);
    final_reduce_kernel<<<1, RED_BLOCK, 0, stream>>>(partials, out);
}